// IntraClassLoss_68015102099647
// MI455X (gfx1250) — compile-verified
//
#include <hip/hip_runtime.h>

typedef __attribute__((ext_vector_type(2))) float v2f;
typedef __attribute__((ext_vector_type(4))) float v4f;
typedef __attribute__((ext_vector_type(8))) float v8f;
typedef __attribute__((ext_vector_type(4))) int   v4i;

#define HW_     (1024 * 1024)
#define BATCH_  16
#define BLOCK_  256
#define ITERS_  8
// pixels per block = BLOCK_*ITERS_*4 = 8192
#define GRIDX_  (HW_ / (BLOCK_ * ITERS_ * 4))   // 128
#define NBLK_   (GRIDX_ * BATCH_)               // 2048

// Stage 1: stream logits+targets (non-temporal, no reuse of the 320 MB),
// per-pixel softmax, accumulate {cnt1..3, S1..3, Q1..3}; wave shfl-reduce,
// then WMMA f32 16x16x4 (B = ones) reduces the 8 wave partials -> ws.
__global__ __launch_bounds__(BLOCK_) void intra_partial_kernel(
    const float* __restrict__ inputs, const int* __restrict__ targets,
    float* __restrict__ partial /* [NBLK_][16] */) {
  const int b = blockIdx.y;
  const int hw_base = blockIdx.x * (BLOCK_ * ITERS_ * 4);
  const float* in0 = inputs + (size_t)b * 4 * HW_;
  const int*   tg  = targets + (size_t)b * HW_;

  // acc[0..2]=cnt, acc[3..5]=S, acc[6..8]=Q for classes 1..3
  float acc[9];
#pragma unroll
  for (int q = 0; q < 9; ++q) acc[q] = 0.0f;

  for (int it = 0; it < ITERS_; ++it) {
    const int hw = hw_base + (it * BLOCK_ + threadIdx.x) * 4;
    // Non-temporal: 320 MB streamed exactly once; don't pollute L2/WGP$.
    v4i t4 = __builtin_nontemporal_load((const v4i*)(tg + hw));
    v4f c0 = __builtin_nontemporal_load((const v4f*)(in0 + (size_t)0 * HW_ + hw));
    v4f c1 = __builtin_nontemporal_load((const v4f*)(in0 + (size_t)1 * HW_ + hw));
    v4f c2 = __builtin_nontemporal_load((const v4f*)(in0 + (size_t)2 * HW_ + hw));
    v4f c3 = __builtin_nontemporal_load((const v4f*)(in0 + (size_t)3 * HW_ + hw));

    const float a0[4] = {c0.x, c0.y, c0.z, c0.w};
    const float a1[4] = {c1.x, c1.y, c1.z, c1.w};
    const float a2[4] = {c2.x, c2.y, c2.z, c2.w};
    const float a3[4] = {c3.x, c3.y, c3.z, c3.w};
    const int   tt[4] = {t4.x, t4.y, t4.z, t4.w};

#pragma unroll
    for (int j = 0; j < 4; ++j) {
      float m = fmaxf(fmaxf(a0[j], a1[j]), fmaxf(a2[j], a3[j]));
      float e0 = __expf(a0[j] - m);
      float e1 = __expf(a1[j] - m);
      float e2 = __expf(a2[j] - m);
      float e3 = __expf(a3[j] - m);
      float inv = 1.0f / (e0 + e1 + e2 + e3);
      int t = tt[j];
      float et = (t == 1) ? e1 : ((t == 2) ? e2 : e3);
      float p = et * inv;
      float pp = p * p;
      bool m1 = (t == 1), m2 = (t == 2), m3 = (t == 3);
      acc[0] += m1 ? 1.0f : 0.0f;
      acc[1] += m2 ? 1.0f : 0.0f;
      acc[2] += m3 ? 1.0f : 0.0f;
      acc[3] += m1 ? p : 0.0f;
      acc[4] += m2 ? p : 0.0f;
      acc[5] += m3 ? p : 0.0f;
      acc[6] += m1 ? pp : 0.0f;
      acc[7] += m2 ? pp : 0.0f;
      acc[8] += m3 ? pp : 0.0f;
    }
  }

  // wave32 butterfly reduce
#pragma unroll
  for (int q = 0; q < 9; ++q) {
#pragma unroll
    for (int off = 16; off > 0; off >>= 1)
      acc[q] += __shfl_xor(acc[q], off, 32);
  }

  __shared__ float P[8][16];  // [wave][quantity], quantities 9..15 zero-pad
  const int lane = threadIdx.x & 31;
  const int wv = threadIdx.x >> 5;
  if (lane == 0) {
#pragma unroll
    for (int q = 0; q < 9; ++q) P[wv][q] = acc[q];
#pragma unroll
    for (int q = 9; q < 16; ++q) P[wv][q] = 0.0f;
  }
  __syncthreads();

  if (threadIdx.x < 32) {  // wave 0, EXEC all ones -> legal WMMA
    const int r = lane & 15;   // A-matrix row M
    const int h = lane >> 4;   // half selects K={0,1} vs K={2,3}
    // A[m][k] = P[k][m]  (16x4 f32 A layout: lanes0-15 K0/K1, lanes16-31 K2/K3)
    v2f a_lo, a_hi, b_one;
    a_lo.x = P[h * 2 + 0][r];
    a_lo.y = P[h * 2 + 1][r];
    a_hi.x = P[h * 2 + 4][r];
    a_hi.y = P[h * 2 + 5][r];
    b_one.x = 1.0f;
    b_one.y = 1.0f;  // B = ones(4x16): layout-invariant
    v8f cz = {};
    // D[m][n] = sum_k P[k][m]  (waves 0..3), then += waves 4..7
    v8f d = __builtin_amdgcn_wmma_f32_16x16x4_f32(
        false, a_lo, false, b_one, (short)0, cz, false, false);
    d = __builtin_amdgcn_wmma_f32_16x16x4_f32(
        false, a_hi, false, b_one, (short)0, d, false, false);

    float* outp = partial + (size_t)(blockIdx.y * gridDim.x + blockIdx.x) * 16;
    if (lane == 0) {  // lane0 holds D[0..7][0] in d[0..7]
#pragma unroll
      for (int q = 0; q < 8; ++q) outp[q] = d[q];
    }
    if (lane == 16) {  // lane16 d[0] = D[8][0] = quantity 8
      outp[8] = d[0];
    }
  }
}

// Stage 2: deterministic reduction of NBLK_ partials + final formula.
__global__ __launch_bounds__(256) void intra_final_kernel(
    const float* __restrict__ partial, float* __restrict__ out) {
  __shared__ float R[8][16];
  float acc[9];
#pragma unroll
  for (int q = 0; q < 9; ++q) acc[q] = 0.0f;

  for (int e = threadIdx.x; e < NBLK_; e += 256) {
    const float* p = partial + (size_t)e * 16;
#pragma unroll
    for (int q = 0; q < 9; ++q) acc[q] += p[q];
  }
#pragma unroll
  for (int q = 0; q < 9; ++q) {
#pragma unroll
    for (int off = 16; off > 0; off >>= 1)
      acc[q] += __shfl_xor(acc[q], off, 32);
  }
  const int lane = threadIdx.x & 31;
  const int wv = threadIdx.x >> 5;
  if (lane == 0) {
#pragma unroll
    for (int q = 0; q < 9; ++q) R[wv][q] = acc[q];
  }
  __syncthreads();

  if (threadIdx.x == 0) {
    float tot[9];
#pragma unroll
    for (int q = 0; q < 9; ++q) tot[q] = 0.0f;
    for (int w = 0; w < 8; ++w)
#pragma unroll
      for (int q = 0; q < 9; ++q) tot[q] += R[w][q];

    const float EPS = 1e-6f;
    float res = 0.0f;
#pragma unroll
    for (int c = 0; c < 3; ++c) {
      float cnt = tot[c], S = tot[3 + c], Q = tot[6 + c];
      float mean = S / (cnt + EPS);
      float var = (Q - 2.0f * mean * S + mean * mean * cnt) / (cnt + EPS);
      if (cnt > 0.0f) res += var;
    }
    out[0] = res * (1.0f / 3.0f);
  }
}

extern "C" void kernel_launch(void* const* d_in, const int* in_sizes, int n_in,
                              void* d_out, int out_size, void* d_ws, size_t ws_size,
                              hipStream_t stream) {
  const float* inputs = (const float*)d_in[0];   // [16,4,1024,1024] f32
  const int* targets = (const int*)d_in[1];      // [16,1024,1024] i32
  float* out = (float*)d_out;                    // scalar f32
  float* partial = (float*)d_ws;                 // NBLK_*16 floats = 128 KB

  dim3 grid(GRIDX_, BATCH_);
  intra_partial_kernel<<<grid, BLOCK_, 0, stream>>>(inputs, targets, partial);
  intra_final_kernel<<<1, 256, 0, stream>>>(partial, out);
}